// Pyramid_Strings_69741678953238
// MI455X (gfx1250) — compile-verified
//
#include <hip/hip_runtime.h>
#include <math.h>

// ---------------- problem constants (fixed by the reference) ----------------
#define BATCH 64
#define IN    512
#define SZ0   256
#define SZ1   64
#define PX0   (BATCH * SZ0 * SZ0)   // 4,194,304 level-0 pixels
#define PX1   (BATCH * SZ1 * SZ1)   //   262,144 level-1 pixels
#define NB0   (PX0 / 512)           // 8192 count/write blocks, level 0
#define NB1   (PX1 / 512)           //  512 count/write blocks, level 1
#define NBALL (NB0 + NB1)           // 8704

typedef __attribute__((ext_vector_type(2))) float v2f;
typedef __attribute__((ext_vector_type(8))) float v8f;

// ---------------------------------------------------------------------------
// K1: bilinear resize 512 -> 256, align_corners=True.
// plane 0: b_edges -> e0 (clipped to [0,1]); planes 1,2: sobel channels -> s0.
// pos = j*511/255 (same f32 arithmetic order as the reference).
// ---------------------------------------------------------------------------
__global__ void k_resize(const float* __restrict__ edges,
                         const float* __restrict__ sobel,
                         float* __restrict__ e0, float* __restrict__ s0) {
  int plane = blockIdx.y;
  int pix   = blockIdx.x * 256 + threadIdx.x;       // b*65536 + y*256 + x
  int b = pix >> 16, rem = pix & 65535, y = rem >> 8, x = rem & 255;

  const float* src;
  if (plane == 0) src = edges + (size_t)b * IN * IN;
  else            src = sobel + (size_t)b * 2 * IN * IN + (size_t)(plane - 1) * IN * IN;

  float py = (float)(y * 511) / 255.0f;
  float px = (float)(x * 511) / 255.0f;
  int ly = (int)py, lx = (int)px;
  float wy = py - (float)ly, wx = px - (float)lx;
  int hy = min(ly + 1, IN - 1), hx = min(lx + 1, IN - 1);

  const float* r0 = src + (size_t)ly * IN;
  const float* r1 = src + (size_t)hy * IN;
  __builtin_prefetch(r1 + lx, 0, 1);                // gfx1250 global_prefetch_b8

  float v00 = r0[lx], v01 = r0[hx], v10 = r1[lx], v11 = r1[hx];
  // interp along y first, then x (matches the reference's expansion)
  float v = (1.f - wx) * ((1.f - wy) * v00 + wy * v10) +
            wx         * ((1.f - wy) * v01 + wy * v11);

  if (plane == 0) { v = fminf(fmaxf(v, 0.f), 1.f); e0[pix] = v; }
  else            s0[(size_t)(plane - 1) * PX0 + pix] = v;
}

// ---------------------------------------------------------------------------
// K2: 4x4 area pool 256 -> 64 (mean of 16), for e0 and both s0 channels.
// ---------------------------------------------------------------------------
__global__ void k_pool(const float* __restrict__ e0, const float* __restrict__ s0,
                       float* __restrict__ e1, float* __restrict__ s1) {
  int plane = blockIdx.y;
  int q = blockIdx.x * 256 + threadIdx.x;           // b*4096 + y*64 + x
  int b = q >> 12, rem = q & 4095, y = rem >> 6, x = rem & 63;

  const float* src = (plane == 0) ? e0 : (s0 + (size_t)(plane - 1) * PX0);
  const float* base = src + (size_t)b * SZ0 * SZ0 + (y * 4) * SZ0 + x * 4;

  float s = 0.f;
  for (int dy = 0; dy < 4; ++dy) {
    float4 v = *(const float4*)(base + dy * SZ0);   // 16B aligned
    s += v.x + v.y + v.z + v.w;
  }
  s *= (1.0f / 16.0f);
  if (plane == 0) e1[q] = s;
  else            s1[(size_t)(plane - 1) * PX1 + q] = s;
}

// ---------------------------------------------------------------------------
// K3: per-block edge counts. 256 threads = 8 waves; each wave reduces 64
// flags with one V_WMMA_F32_16X16X4_F32:
//   A(16x4) = flags (any slot<->pixel bijection), B(4x16) = all ones
//   => D[m][n] = rowsum_A(m);  sum of all 256 D entries = 16 * count.
// This is layout-invariant (all-ones B, full-D sum). Exact in f32 (<=1024).
// EXEC is all ones: no divergence before the WMMA.
// ---------------------------------------------------------------------------
__global__ void k_count(const float* __restrict__ e0, const float* __restrict__ e1,
                        int* __restrict__ blockSums) {
  int bid = blockIdx.x;
  int tid = threadIdx.x, wave = tid >> 5, lane = tid & 31;

  const float* src;
  int base;
  if (bid < NB0) { src = e0; base = bid * 512 + wave * 64; }
  else           { src = e1; base = (bid - NB0) * 512 + wave * 64; }

  float2 v = *((const float2*)src + (base >> 1) + lane);  // 2 contiguous pixels/lane

  v2f a; a[0] = (v.x != 0.f) ? 1.f : 0.f; a[1] = (v.y != 0.f) ? 1.f : 0.f;
  v2f bo; bo[0] = 1.f; bo[1] = 1.f;
  v8f c = {};
  v8f d = __builtin_amdgcn_wmma_f32_16x16x4_f32(false, a, false, bo,
                                                (short)0, c, false, false);
  float s = d[0] + d[1] + d[2] + d[3] + d[4] + d[5] + d[6] + d[7];
  for (int off = 16; off > 0; off >>= 1) s += __shfl_xor(s, off, 32);
  s *= (1.0f / 16.0f);                               // exact: integer <= 64

  __shared__ float wsum[8];
  if (lane == 0) wsum[wave] = s;
  __syncthreads();
  if (tid == 0) {
    float t = 0.f;
    for (int w = 0; w < 8; ++w) t += wsum[w];
    blockSums[bid] = (int)(t + 0.5f);
  }
}

// ---------------------------------------------------------------------------
// K4: single-block exclusive scan over the 8704 block sums (level-0 blocks
// first, then level-1 => scanned offsets ARE the global row order).
// ---------------------------------------------------------------------------
__global__ void k_scan(const int* __restrict__ sums, int* __restrict__ offs, int nb) {
  __shared__ int tmp[256];
  __shared__ int carryS;
  int tid = threadIdx.x;
  if (tid == 0) carryS = 0;
  __syncthreads();
  for (int chunk = 0; chunk < nb; chunk += 256) {
    int i = chunk + tid;
    int v = (i < nb) ? sums[i] : 0;
    tmp[tid] = v;
    __syncthreads();
    for (int st = 1; st < 256; st <<= 1) {          // Hillis-Steele inclusive
      int add = (tid >= st) ? tmp[tid - st] : 0;
      __syncthreads();
      tmp[tid] += add;
      __syncthreads();
    }
    int incl = tmp[tid];
    int c = carryS;
    if (i < nb) offs[i] = c + incl - v;             // exclusive + carry
    __syncthreads();
    if (tid == 255) carryS = c + incl;
    __syncthreads();
  }
}

// ---------------------------------------------------------------------------
// K5: compaction write. Same 512-pixel blocks as K3; in-block rank follows
// flat pixel order (two 256-pixel passes, wave32 ballot prefix).
// p_lin is staged bit-wise into out[10N..11N); resolved by K6.
// ---------------------------------------------------------------------------
__global__ void k_write(const float* __restrict__ e0, const float* __restrict__ s0,
                        const float* __restrict__ e1, const float* __restrict__ s1,
                        const int* __restrict__ offs, float* __restrict__ out,
                        int* __restrict__ table, int N) {
  int bid = blockIdx.x, tid = threadIdx.x;
  int wave = tid >> 5, lane = tid & 31;
  int level, base;
  if (bid < NB0) { level = 0; base = bid * 512; }
  else           { level = 1; base = (bid - NB0) * 512; }

  __shared__ int wcnt[8];
  __shared__ int passBaseS;
  if (tid == 0) passBaseS = offs[bid];
  __syncthreads();

  for (int pass = 0; pass < 2; ++pass) {
    int pp = base + pass * 256 + tid;               // pixel within level
    float ev = (level == 0) ? e0[pp] : e1[pp];
    bool flag = (ev != 0.f);

    unsigned mask = (unsigned)__ballot((int)flag);  // wave32: low 32 bits
    if (lane == 0) wcnt[wave] = __popc(mask);
    int lanePre = __popc(mask & ((1u << lane) - 1u));
    __syncthreads();
    int wbase = 0;
    for (int w = 0; w < wave; ++w) wbase += wcnt[w];
    int r = passBaseS + wbase + lanePre;            // global row id

    if (flag) {
      int b_, y_, x_, my_lin, p_lin;
      float cy, cx, nx, ny, sizev;
      if (level == 0) {
        b_ = pp >> 16; int rem = pp & 65535; y_ = rem >> 8; x_ = rem & 255;
        my_lin = pp;
        p_lin = PX0 + b_ * 4096 + (y_ >> 2) * 64 + (x_ >> 2);  // nearest-up parent
        sizev = 1.f; cy = (float)y_ + 0.5f; cx = (float)x_ + 0.5f;
        nx = s0[pp]; ny = s0[PX0 + pp];
      } else {
        b_ = pp >> 12; int rem = pp & 4095; y_ = rem >> 6; x_ = rem & 63;
        my_lin = PX0 + pp;
        p_lin = my_lin;                                         // last level: self
        sizev = 4.f; cy = ((float)y_ + 0.5f) * 4.f; cx = ((float)x_ + 0.5f) * 4.f;
        nx = s1[pp]; ny = s1[PX1 + pp];
      }
      float len = sqrtf(nx * nx + ny * ny);
      nx /= len; ny /= len;
      float h = sizev * 0.5f;
      float rty = ny * h, rtx = -nx * h;            // cross([n,0],[0,0,1])

      out[2 * r]             = cy + rty;  out[2 * r + 1]             = cx + rtx;  // locs_lf
      out[2 * N + 2 * r]     = cy - rty;  out[2 * N + 2 * r + 1]     = cx - rtx;  // locs_rt
      out[4 * N + 2 * r]     = nx;        out[4 * N + 2 * r + 1]     = ny;        // norms
      out[6 * N + r]         = sizev;                                             // sizes
      out[7 * N + 2 * r]     = cy;        out[7 * N + 2 * r + 1]     = cx;        // centers
      out[9 * N + r]         = (float)b_;                                         // imgid
      ((int*)out)[10 * N + r] = p_lin;     // staged bit-wise, resolved in K6
      out[11 * N + r]        = (float)my_lin;                                     // my_lin
      table[my_lin] = r;                   // scatter for parent lookup
    }
    __syncthreads();
    if (tid == 0) {
      int t = 0;
      for (int w = 0; w < 8; ++w) t += wcnt[w];
      passBaseS += t;
    }
    __syncthreads();
  }
}

// ---------------------------------------------------------------------------
// K6: p_rowids = table[p_lin] (parents are always edge pixels at their level,
// since e1 is a mean of non-negative values containing the child's e0 value).
// ---------------------------------------------------------------------------
__global__ void k_resolve(float* __restrict__ out, const int* __restrict__ table, int N) {
  int r = blockIdx.x * 256 + threadIdx.x;
  if (r >= N) return;
  int pl = ((const int*)out)[10 * N + r];
  out[10 * N + r] = (float)table[pl];
}

// ---------------------------------------------------------------------------
extern "C" void kernel_launch(void* const* d_in, const int* in_sizes, int n_in,
                              void* d_out, int out_size, void* d_ws, size_t ws_size,
                              hipStream_t stream) {
  const float* edges = (const float*)d_in[0];   // (64,1,512,512)
  const float* sobel = (const float*)d_in[1];   // (64,2,512,512)
  float* out = (float*)d_out;
  int N = out_size / 12;                        // 12 floats per compacted row

  // workspace layout (all 4-byte elements; ~69 MiB total)
  float* ws = (float*)d_ws;
  float* e0 = ws;                               // PX0
  float* s0 = ws + (size_t)PX0;                 // 2*PX0
  float* e1 = ws + (size_t)3 * PX0;             // PX1
  float* s1 = e1 + (size_t)PX1;                 // 2*PX1
  int* blockSums = (int*)(s1 + (size_t)2 * PX1);// NBALL
  int* offs      = blockSums + NBALL;           // NBALL
  int* table     = offs + NBALL;                // PX0+PX1

  k_resize<<<dim3(PX0 / 256, 3), 256, 0, stream>>>(edges, sobel, e0, s0);
  k_pool  <<<dim3(PX1 / 256, 3), 256, 0, stream>>>(e0, s0, e1, s1);
  k_count <<<NBALL, 256, 0, stream>>>(e0, e1, blockSums);
  k_scan  <<<1, 256, 0, stream>>>(blockSums, offs, NBALL);
  k_write <<<NBALL, 256, 0, stream>>>(e0, s0, e1, s1, offs, out, table, N);
  if (N > 0) k_resolve<<<(N + 255) / 256, 256, 0, stream>>>(out, table, N);
}